// GAT_Encoder_18322330485334
// MI455X (gfx1250) — compile-verified
//
#include <hip/hip_runtime.h>

typedef float v2f __attribute__((ext_vector_type(2)));
typedef float v8f __attribute__((ext_vector_type(8)));

#define NNODES 30000
#define HEADS 4
#define SLOPE 0.2f
#define KCHUNK 128           // K-rows of the B strip staged in LDS per step
#define PAIRSTRIDE 160       // dwords between k-row-pairs in LDS (bank-conflict-free)

// ---------------------------------------------------------------------------
// GEMM: C[M,Nout] = A[M,K] * B[K,Nout], fp32, via V_WMMA_F32_16X16X4_F32.
// Block = 256 threads = 8 waves; block owns a 64-wide N strip and 8 M-tiles.
// B chunk (128 x 64) is staged in LDS in *fragment order*:
//   element (k, n) -> ldsB[(k>>1)*PAIRSTRIDE + n*2 + (k&1)]
// so each lane's B fragment {B[ka][n], B[ka+1][n]} is one aligned ds_load_b64.
// Fragment layouts per CDNA5 ISA:
//   A 16x4 : lane l -> row M=l%16, VGPR v -> K = 2*(l/16)+v
//   B 4x16 : lane l -> col N=l%16, VGPR v -> K = 2*(l/16)+v
//   C/D    : lane l, VGPR v -> M = v + 8*(l/16), N = l%16
// ---------------------------------------------------------------------------
__global__ __launch_bounds__(256)
void gemm_xw_wmma(const float* __restrict__ A, const float* __restrict__ B,
                  float* __restrict__ C, int M, int K, int Nout) {
  __shared__ float ldsB[(KCHUNK / 2) * PAIRSTRIDE];   // 40 KB

  const int tid  = threadIdx.x;
  const int lane = tid & 31;
  const int wave = tid >> 5;
  const int Mtiles = M >> 4;

  const int n0 = blockIdx.y * 64;
  int mt = blockIdx.x * 8 + wave;
  const bool active = (mt < Mtiles);
  if (!active) mt = Mtiles - 1;            // wave-uniform clamp: loads stay safe,
                                           // EXEC stays all-1s for WMMA
  const int m0   = mt * 16;
  const int half = lane >> 4;
  const int lm   = lane & 15;

  // cooperative-load coordinates for the B strip
  const int lrow = tid >> 4;               // 0..15
  const int lcol = (tid & 15) * 4;         // 0,4,...,60

  v8f acc0 = {}, acc1 = {}, acc2 = {}, acc3 = {};
  const float* __restrict__ arow = A + (size_t)(m0 + lm) * K;

  for (int kb = 0; kb < K; kb += KCHUNK) {
    __syncthreads();                       // previous chunk's compute done
#pragma unroll
    for (int r = 0; r < KCHUNK; r += 16) {
      const int k = r + lrow;
      const float4 v =
          *(const float4*)(B + (size_t)(kb + k) * Nout + n0 + lcol);
      float* p = &ldsB[(k >> 1) * PAIRSTRIDE + lcol * 2 + (k & 1)];
      p[0] = v.x; p[2] = v.y; p[4] = v.z; p[6] = v.w;
    }
    __syncthreads();                       // chunk visible to all waves

    if (kb + KCHUNK < K)                   // prefetch next A chunk for this row
      __builtin_prefetch(arow + kb + KCHUNK, 0, 0);

#pragma unroll 4
    for (int k = 0; k < KCHUNK; k += 4) {
      v2f a = *(const v2f*)(arow + kb + k + 2 * half);
      // pair-row (k/2 + half), column lm (+16 per N sub-tile)
      const float* __restrict__ bp =
          &ldsB[((k >> 1) + half) * PAIRSTRIDE + lm * 2];
      v2f b;
      b = *(const v2f*)(bp + 0);
      acc0 = __builtin_amdgcn_wmma_f32_16x16x4_f32(false, a, false, b, (short)0, acc0, false, false);
      b = *(const v2f*)(bp + 32);
      acc1 = __builtin_amdgcn_wmma_f32_16x16x4_f32(false, a, false, b, (short)0, acc1, false, false);
      b = *(const v2f*)(bp + 64);
      acc2 = __builtin_amdgcn_wmma_f32_16x16x4_f32(false, a, false, b, (short)0, acc2, false, false);
      b = *(const v2f*)(bp + 96);
      acc3 = __builtin_amdgcn_wmma_f32_16x16x4_f32(false, a, false, b, (short)0, acc3, false, false);
    }
  }

  if (active) {
    float* __restrict__ crow = C + (size_t)(m0 + 8 * half) * Nout + n0 + lm;
#pragma unroll
    for (int v = 0; v < 8; ++v) {
      crow[(size_t)v * Nout + 0]  = acc0[v];
      crow[(size_t)v * Nout + 16] = acc1[v];
      crow[(size_t)v * Nout + 32] = acc2[v];
      crow[(size_t)v * Nout + 48] = acc3[v];
    }
  }
}

// ---------------------------------------------------------------------------
// Attention logits: als[n,h] = <H[n,h,:], a_src[h,:]>, ald likewise.
// ---------------------------------------------------------------------------
__global__ void attn_logits(const float* __restrict__ H, const float* __restrict__ a_s,
                            const float* __restrict__ a_d, float* __restrict__ als,
                            float* __restrict__ ald, int Nn, int C) {
  int i = blockIdx.x * blockDim.x + threadIdx.x;
  if (i >= Nn * HEADS) return;
  int n = i / HEADS, h = i % HEADS;
  const float* hp = H + (size_t)n * HEADS * C + (size_t)h * C;
  float s = 0.f, d = 0.f;
  for (int c = 0; c < C; ++c) { float v = hp[c]; s += v * a_s[h * C + c]; d += v * a_d[h * C + c]; }
  als[i] = s; ald[i] = d;
}

__global__ void zero_f(float* __restrict__ p, long long n) {
  long long i = (long long)blockIdx.x * blockDim.x + threadIdx.x;
  if (i < n) p[i] = 0.f;
}

// monotonic uint key for float atomic-max (handles negatives); key 0 < all reals
__device__ __forceinline__ unsigned fkey(float x) {
  unsigned b = __float_as_uint(x);
  return (b & 0x80000000u) ? ~b : (b | 0x80000000u);
}
__device__ __forceinline__ float funkey(unsigned k) {
  unsigned b = (k & 0x80000000u) ? (k & 0x7FFFFFFFu) : ~k;
  return __uint_as_float(b);
}

// pass 1: e = leaky_relu(als[src]+ald[dst]); segment max over dst
__global__ void edge_scores(const int* __restrict__ src, const int* __restrict__ dst,
                            int E, const float* __restrict__ als, const float* __restrict__ ald,
                            float* __restrict__ ews, unsigned* __restrict__ mkey) {
  int e = blockIdx.x * blockDim.x + threadIdx.x;
  const int EL = E + NNODES;
  if (e >= EL) return;
  int s = (e < E) ? src[e] : (e - E);
  int d = (e < E) ? dst[e] : (e - E);
#pragma unroll
  for (int h = 0; h < HEADS; ++h) {
    float v = als[s * HEADS + h] + ald[d * HEADS + h];
    v = (v > 0.f) ? v : SLOPE * v;
    ews[(size_t)e * HEADS + h] = v;
    atomicMax(&mkey[d * HEADS + h], fkey(v));
  }
}

// pass 2: ex = exp(e - m[dst]); segment sum into denom
__global__ void edge_exp(const int* __restrict__ src, const int* __restrict__ dst,
                         int E, float* __restrict__ ews,
                         const unsigned* __restrict__ mkey, float* __restrict__ den) {
  int e = blockIdx.x * blockDim.x + threadIdx.x;
  const int EL = E + NNODES;
  if (e >= EL) return;
  int d = (e < E) ? dst[e] : (e - E);
#pragma unroll
  for (int h = 0; h < HEADS; ++h) {
    float m = funkey(mkey[d * HEADS + h]);
    float ex = __expf(ews[(size_t)e * HEADS + h] - m);
    ews[(size_t)e * HEADS + h] = ex;
    atomicAdd(&den[d * HEADS + h], ex);
  }
}

// pass 3: agg[dst] += alpha * H[src], 4 channels per thread
__global__ void edge_aggregate(const int* __restrict__ src, const int* __restrict__ dst,
                               int E, const float* __restrict__ H,
                               const float* __restrict__ ews, const float* __restrict__ den,
                               float* __restrict__ agg, int C) {
  const int Fout = HEADS * C;
  const int chunks = Fout >> 2;
  long long tid = (long long)blockIdx.x * blockDim.x + threadIdx.x;
  long long total = (long long)(E + NNODES) * chunks;
  if (tid >= total) return;
  int e = (int)(tid / chunks);
  int j = (int)(tid % chunks);
  int f = j * 4;
  int h = f / C;
  int s = (e < E) ? src[e] : (e - E);
  int d = (e < E) ? dst[e] : (e - E);
  float alpha = ews[(size_t)e * HEADS + h] / (den[d * HEADS + h] + 1e-16f);
  const float4 hv = *(const float4*)(H + (size_t)s * Fout + f);
  float* op = agg + (size_t)d * Fout + f;
  atomicAdd(op + 0, hv.x * alpha);
  atomicAdd(op + 1, hv.y * alpha);
  atomicAdd(op + 2, hv.z * alpha);
  atomicAdd(op + 3, hv.w * alpha);
}

// concat layers: out = relu(agg + b)   (safe in place)
__global__ void bias_relu(const float* __restrict__ agg, const float* __restrict__ b,
                          float* __restrict__ out, int Nn, int F) {
  int i = blockIdx.x * blockDim.x + threadIdx.x;
  if (i >= Nn * F) return;
  float v = agg[i] + b[i % F];
  out[i] = v > 0.f ? v : 0.f;
}

// mean layers: out[n,c] = mean_h(agg[n,h,c]) + b[c]
__global__ void mean_bias(const float* __restrict__ agg, const float* __restrict__ b,
                          float* __restrict__ out, int Nn, int C) {
  int i = blockIdx.x * blockDim.x + threadIdx.x;
  if (i >= Nn * C) return;
  int n = i / C, c = i % C;
  const float* p = agg + (size_t)n * HEADS * C + c;
  out[i] = 0.25f * (p[0] + p[C] + p[2 * C] + p[3 * C]) + b[c];
}

// ---------------------------------------------------------------------------
static void run_gat_layer(hipStream_t st, const float* in, int Fin, int C,
                          const float* W, const float* a_s, const float* a_d, const float* b,
                          float* Hbuf, float* agg, float* outMean,
                          const int* src, const int* dst, int E,
                          float* ALS, float* ALD, float* EWS, unsigned* MKEY, float* DEN) {
  const int Nn = NNODES;
  const int Fout = HEADS * C;
  // GEMM: grid = (M-tile groups of 8) x (64-wide N strips)
  const int Mtiles = Nn / 16;
  dim3 ggrid((Mtiles + 7) / 8, Fout / 64);
  gemm_xw_wmma<<<ggrid, 256, 0, st>>>(in, W, Hbuf, Nn, Fin, Fout);
  // logits
  int t = Nn * HEADS;
  attn_logits<<<(t + 255) / 256, 256, 0, st>>>(Hbuf, a_s, a_d, ALS, ALD, Nn, C);
  // clear segment buffers
  zero_f<<<(t + 255) / 256, 256, 0, st>>>(DEN, t);
  zero_f<<<(t + 255) / 256, 256, 0, st>>>((float*)MKEY, t);
  long long nagg = (long long)Nn * Fout;
  zero_f<<<(int)((nagg + 255) / 256), 256, 0, st>>>(agg, nagg);
  // edge passes
  const int EL = E + Nn;
  edge_scores<<<(EL + 255) / 256, 256, 0, st>>>(src, dst, E, ALS, ALD, EWS, MKEY);
  edge_exp<<<(EL + 255) / 256, 256, 0, st>>>(src, dst, E, EWS, MKEY, DEN);
  long long tot = (long long)EL * (Fout >> 2);
  edge_aggregate<<<(int)((tot + 255) / 256), 256, 0, st>>>(src, dst, E, Hbuf, EWS, DEN, agg, C);
  // finalize
  if (outMean) {
    mean_bias<<<(Nn * C + 255) / 256, 256, 0, st>>>(agg, b, outMean, Nn, C);
  } else {
    bias_relu<<<(int)((nagg + 255) / 256), 256, 0, st>>>(agg, b, agg, Nn, Fout);
  }
}

extern "C" void kernel_launch(void* const* d_in, const int* in_sizes, int n_in,
                              void* d_out, int out_size, void* d_ws, size_t ws_size,
                              hipStream_t stream) {
  const float* x   = (const float*)d_in[0];
  const int*   ei  = (const int*)d_in[1];
  const int E = in_sizes[1] / 2;
  const int* src = ei;
  const int* dst = ei + E;
  const float *W1 = (const float*)d_in[2],  *as1 = (const float*)d_in[3],
              *ad1 = (const float*)d_in[4], *b1  = (const float*)d_in[5];
  const float *W2 = (const float*)d_in[6],  *as2 = (const float*)d_in[7],
              *ad2 = (const float*)d_in[8], *b2  = (const float*)d_in[9];
  const float *Wm = (const float*)d_in[10], *asm_ = (const float*)d_in[11],
              *adm = (const float*)d_in[12], *bm = (const float*)d_in[13];
  const float *Wl = (const float*)d_in[14], *asl = (const float*)d_in[15],
              *adl = (const float*)d_in[16], *bl = (const float*)d_in[17];

  // workspace arena
  char* ws = (char*)d_ws;
  size_t off = 0;
  auto alloc = [&](size_t bytes) -> void* {
    void* p = ws + off;
    off += (bytes + 255) & ~(size_t)255;
    return p;
  };
  float*    Hbuf = (float*)alloc((size_t)NNODES * 512 * 4);  // h = X@W (max 512 wide)
  float*    A0   = (float*)alloc((size_t)NNODES * 256 * 4);  // act1 / agg for head layers
  float*    A1   = (float*)alloc((size_t)NNODES * 512 * 4);  // act2
  float*    ALS  = (float*)alloc((size_t)NNODES * HEADS * 4);
  float*    ALD  = (float*)alloc((size_t)NNODES * HEADS * 4);
  float*    EWS  = (float*)alloc((size_t)(E + NNODES) * HEADS * 4);
  unsigned* MKEY = (unsigned*)alloc((size_t)NNODES * HEADS * 4);
  float*    DEN  = (float*)alloc((size_t)NNODES * HEADS * 4);

  float* zmean   = (float*)d_out;                    // [30000, 32]
  float* zlogstd = (float*)d_out + (size_t)NNODES * 32;

  // layer 1: 256 -> 4x64 concat (=256), relu; agg+act in A0
  run_gat_layer(stream, x, 256, 64, W1, as1, ad1, b1, Hbuf, A0, nullptr,
                src, dst, E, ALS, ALD, EWS, MKEY, DEN);
  // layer 2: 256 -> 4x128 concat (=512), relu; agg+act in A1
  run_gat_layer(stream, A0, 256, 128, W2, as2, ad2, b2, Hbuf, A1, nullptr,
                src, dst, E, ALS, ALD, EWS, MKEY, DEN);
  // z_mean: 512 -> 4x32 mean (=32); agg reuses A0
  run_gat_layer(stream, A1, 512, 32, Wm, asm_, adm, bm, Hbuf, A0, zmean,
                src, dst, E, ALS, ALD, EWS, MKEY, DEN);
  // z_logstd: 512 -> 4x32 mean (=32); agg reuses A0
  run_gat_layer(stream, A1, 512, 32, Wl, asl, adl, bl, Hbuf, A0, zlogstd,
                src, dst, E, ALS, ALD, EWS, MKEY, DEN);
}